// CenterLoss_5738076307838
// MI455X (gfx1250) — compile-verified
//
#include <hip/hip_runtime.h>

// CenterLoss on MI455X (gfx1250, wave32).
//
// Only distmat[i, labels[i]] is used by the reference, so the real work is
//   dist_i = ||x_i - c_{labels[i]}||^2, clamp to [1e-12, 1e12], mean.
// Mandatory HBM traffic: x (16 MB, streamed once, non-temporal) + gathered
// center rows (~16 MB; the 16 MB centers table lives in the 192 MB L2)
// => ~32 MB / 23.3 TB/s ~= 1.4 us floor. Compute is ~free, so we organize
// around wide parallel loads and use V_WMMA_F32_16X16X4_F32 on the
// DIFFERENCE d = x - c: per-lane A(row)/B(col) layouts coincide, so one
// register pair feeds both operands and diag(D) = ||x_m - c_m||^2 directly
// (no separate norm FMAs, no operand repacking).
//
// Occupancy: K=512 is split 4 ways (4 waves per 16-sample tile, 128 K each)
// -> 2048 waves in flight for latency hiding on the gather stream; partial
// diagonals are combined in LDS. All reductions are fixed-order
// (deterministic across graph replays; no float atomics).

typedef __attribute__((ext_vector_type(2))) float v2f;
typedef __attribute__((ext_vector_type(8))) float v8f;

#define BATCH   8192
#define DIM     512
#define SPLIT   4                          // K-split: waves per 16-sample tile
#define KCHUNK  (DIM / SPLIT)              // 128
#define WAVES_PER_BLOCK 8
#define TILES_PER_BLOCK (WAVES_PER_BLOCK / SPLIT)   // 2
#define NTILES  (BATCH / 16)               // 512
#define NBLOCKS (NTILES / TILES_PER_BLOCK) // 256

__global__ __launch_bounds__(256) void center_dist_kernel(
    const float* __restrict__ x,
    const long long* __restrict__ labels,      // int64 labels
    const float* __restrict__ centers,
    float* __restrict__ partials)              // [NBLOCKS]
{
    const int lane   = threadIdx.x & 31;
    const int wave   = threadIdx.x >> 5;
    const int tile   = blockIdx.x * TILES_PER_BLOCK + (wave / SPLIT);
    const int kslice = wave & (SPLIT - 1);

    // WMMA f32 16x16x4 operand layout (wave32):
    //   lane L<16  supplies row/col m=L,    K = {k, k+1}
    //   lane L>=16 supplies row/col m=L-16, K = {k+2, k+3}
    const int m      = lane & 15;
    const int koff   = (lane >> 4) << 1;       // 0 or 2
    const int sample = tile * 16 + m;

    const long long lab = labels[sample];
    const float* xrow = x       + (size_t)sample * DIM + kslice * KCHUNK;
    const float* crow = centers + (size_t)lab    * DIM + kslice * KCHUNK;

    v8f acc0 = {};
    v8f acc1 = {};

    // d = x - c serves as BOTH the A and B operand; diag accumulates ||d||^2.
    // Two interleaved accumulator chains hide WMMA D->C latency.
    #pragma unroll 4
    for (int k = 0; k < KCHUNK; k += 8) {
        v2f a0 = __builtin_nontemporal_load((const v2f*)(xrow + k + koff));
        v2f b0 = *(const v2f*)(crow + k + koff);
        v2f a1 = __builtin_nontemporal_load((const v2f*)(xrow + k + 4 + koff));
        v2f b1 = *(const v2f*)(crow + k + 4 + koff);

        v2f d0 = a0 - b0;
        v2f d1 = a1 - b1;

        acc0 = __builtin_amdgcn_wmma_f32_16x16x4_f32(
            false, d0, false, d0, (short)0, acc0, false, false);
        acc1 = __builtin_amdgcn_wmma_f32_16x16x4_f32(
            false, d1, false, d1, (short)0, acc1, false, false);
    }

    // Diagonal of the 16x16 f32 accumulator:
    //   sample s<8  -> VGPR s,   lane s
    //   sample s>=8 -> VGPR s-8, lane s+16
    const int idx = (lane < 8) ? lane : ((lane >= 24) ? (lane - 24) : -1);
    float part = 0.0f;
    #pragma unroll
    for (int r = 0; r < 8; ++r) {
        float d = acc0[r] + acc1[r];
        part = (idx == r) ? d : part;
    }

    // Stash this wave's 16 partial squared-distances, indexed by sample.
    __shared__ float tl[WAVES_PER_BLOCK][16];
    const int s = (lane < 8) ? lane : (lane - 16);   // 0..15 for valid lanes
    if (idx >= 0) tl[wave][s] = part;
    __syncthreads();

    // Wave 0 combines the SPLIT partials per sample (2 tiles * 16 = 32 slots),
    // clamps, and does a fixed-order shuffle reduction.
    if (wave == 0) {
        const int t  = lane >> 4;        // tile-in-block
        const int ss = lane & 15;        // sample-in-tile
        float dsum = 0.0f;
        #pragma unroll
        for (int j = 0; j < SPLIT; ++j) dsum += tl[t * SPLIT + j][ss];
        float dist = fminf(fmaxf(dsum, 1e-12f), 1e12f);   // reference clamp
        #pragma unroll
        for (int off = 16; off > 0; off >>= 1)
            dist += __shfl_xor(dist, off, 32);
        if (lane == 0) partials[blockIdx.x] = dist;
    }
}

// Single-block deterministic final reduction: mean = sum(partials) / BATCH.
__global__ __launch_bounds__(NBLOCKS) void final_reduce_kernel(
    const float* __restrict__ partials, float* __restrict__ out)
{
    __shared__ float buf[NBLOCKS];
    const int t = threadIdx.x;
    buf[t] = partials[t];
    __syncthreads();
    #pragma unroll
    for (int s = NBLOCKS / 2; s > 0; s >>= 1) {
        if (t < s) buf[t] += buf[t + s];
        __syncthreads();
    }
    if (t == 0) out[0] = buf[0] * (1.0f / (float)BATCH);
}

extern "C" void kernel_launch(void* const* d_in, const int* in_sizes, int n_in,
                              void* d_out, int out_size, void* d_ws, size_t ws_size,
                              hipStream_t stream) {
    const float*     x       = (const float*)d_in[0];
    const long long* labels  = (const long long*)d_in[1];   // int64
    const float*     centers = (const float*)d_in[2];
    float* out      = (float*)d_out;
    float* partials = (float*)d_ws;                          // NBLOCKS floats

    center_dist_kernel<<<NBLOCKS, WAVES_PER_BLOCK * 32, 0, stream>>>(
        x, labels, centers, partials);
    final_reduce_kernel<<<1, NBLOCKS, 0, stream>>>(partials, out);
}